// LinearAttentionLayer_89111981457418
// MI455X (gfx1250) — compile-verified
//
#include <hip/hip_runtime.h>
#include <hip/hip_bf16.h>

typedef float v2f __attribute__((ext_vector_type(2)));
typedef float v8f __attribute__((ext_vector_type(8)));

constexpr int kSeq = 8192;
constexpr int kDM  = 1024;   // d_model
constexpr int kNH  = 16;     // heads
constexpr int kDH  = 64;     // head dim
constexpr int kN   = 1024;   // GEMM N (all 4 GEMMs are 8192x1024x1024)
constexpr int kK   = 1024;   // GEMM K
constexpr int kChunk = 128;
constexpr int kNCh   = kSeq / kChunk;   // 64 chunks

__device__ __forceinline__ float act_elu1(float x) {
  // elu(x)+1 = x>0 ? x+1 : exp(x)
  return x > 0.f ? x + 1.f : __expf(x);
}

__device__ __forceinline__ float gelu_tanh(float x) {
  // jax.nn.gelu default (approximate=True)
  const float c0 = 0.7978845608028654f, c1 = 0.044715f;
  float x3 = x * x * x;
  return 0.5f * x * (1.f + tanhf(c0 * (x + c1 * x3)));
}

// C(8192x1024) = A(8192x1024) x W^T, W row-major (out=1024, in=1024).
// Each wave computes a 64x64 tile as 4x4 WMMA 16x16 tiles, K stepped by 4.
// MODE 0: store raw
// MODE 1: store gelu(x + bias)
// MODE 2: store x + bias
// MODE 3: per-row head reduction: C[t, head] = sum_d act(x[t, head*64+d]); ld = kNH
template <int MODE>
__global__ __launch_bounds__(256)
void gemm_f32_wmma(const float* __restrict__ A,
                   const float* __restrict__ W,
                   const float* __restrict__ bias,
                   float* __restrict__ C) {
  const int wave  = (blockIdx.x * blockDim.x + threadIdx.x) >> 5;
  const int tileM = wave >> 4;   // 0..127  (8192/64)
  const int tileN = wave & 15;   // 0..15   (1024/64) == head index for MODE 3
  const int m0 = tileM * 64, n0 = tileN * 64;
  const int lane = threadIdx.x & 31;
  const int half = lane >> 4;    // K pair select per ISA f32 A/B layout
  const int l16  = lane & 15;    // M (A) / N (B) row within tile

  const float* aP[4];
  const float* wP[4];
#pragma unroll
  for (int i = 0; i < 4; ++i) {
    aP[i] = A + (size_t)(m0 + i * 16 + l16) * kK + 2 * half;
    wP[i] = W + (size_t)(n0 + i * 16 + l16) * kK + 2 * half;
  }

  v8f acc[4][4] = {};

  for (int k = 0; k < kK; k += 4) {
    v2f av[4], bv[4];
#pragma unroll
    for (int i = 0; i < 4; ++i) av[i] = *(const v2f*)(aP[i] + k);
#pragma unroll
    for (int i = 0; i < 4; ++i) bv[i] = *(const v2f*)(wP[i] + k);
#pragma unroll
    for (int mt = 0; mt < 4; ++mt)
#pragma unroll
      for (int nt = 0; nt < 4; ++nt)
        acc[mt][nt] = __builtin_amdgcn_wmma_f32_16x16x4_f32(
            false, av[mt], false, bv[nt], (short)0, acc[mt][nt], false, false);
  }

  if (MODE == 3) {
    // act + reduce the wave's 64 columns (exactly one head) per output row
#pragma unroll
    for (int mt = 0; mt < 4; ++mt) {
#pragma unroll
      for (int j = 0; j < 8; ++j) {
        float s = 0.f;
#pragma unroll
        for (int nt = 0; nt < 4; ++nt) s += act_elu1(acc[mt][nt][j]);
        // reduce across the 16 lanes of each half (masks < 16 keep halves apart)
#pragma unroll
        for (int m = 1; m < 16; m <<= 1) s += __shfl_xor(s, m, 32);
        if (l16 == 0) {
          int row = m0 + mt * 16 + 8 * half + j;  // C/D layout: VGPR j = row j (+8 for hi half)
          C[(size_t)row * kNH + tileN] = s;
        }
      }
    }
  } else {
#pragma unroll
    for (int nt = 0; nt < 4; ++nt) {
      float b = (MODE == 0) ? 0.f : bias[n0 + nt * 16 + l16];
#pragma unroll
      for (int mt = 0; mt < 4; ++mt) {
#pragma unroll
        for (int j = 0; j < 8; ++j) {
          float x = acc[mt][nt][j] + b;
          if (MODE == 1) x = gelu_tanh(x);
          C[(size_t)(m0 + mt * 16 + 8 * half + j) * kN + (n0 + nt * 16 + l16)] = x;
        }
      }
    }
  }
}

// ---- scan: out[t,h,v] = cumsum(a*v)/cumsum(a), chunked 2-pass prefix ----

__global__ void scan_chunk_sums(const float* __restrict__ Amat,
                                const float* __restrict__ V,
                                float* __restrict__ csum,
                                float* __restrict__ cden) {
  const int c = blockIdx.x, h = blockIdx.y, v = threadIdx.x;  // 64 threads
  const int t0 = c * kChunk;
  float sv = 0.f, sa = 0.f;
  for (int i = 0; i < kChunk; ++i) {
    int t = t0 + i;
    float a = Amat[(size_t)t * kNH + h];
    sv += a * V[(size_t)t * kN + h * kDH + v];
    sa += a;
  }
  csum[((size_t)c * kNH + h) * kDH + v] = sv;
  if (v == 0) cden[(size_t)c * kNH + h] = sa;
}

__global__ void scan_prefix(float* __restrict__ csum, float* __restrict__ cden) {
  const int tid = threadIdx.x;  // 1024 threads, one per (h, v)
  const int h = tid / kDH, v = tid % kDH;
  float run = 0.f;
  for (int c = 0; c < kNCh; ++c) {
    size_t idx = ((size_t)c * kNH + h) * kDH + v;
    float t = csum[idx];
    csum[idx] = run;   // exclusive prefix
    run += t;
  }
  if (tid < kNH) {
    float r = 0.f;
    for (int c = 0; c < kNCh; ++c) {
      size_t idx = (size_t)c * kNH + tid;
      float t = cden[idx];
      cden[idx] = r;
      r += t;
    }
  }
}

__global__ void scan_apply(const float* __restrict__ Amat,
                           const float* __restrict__ V,
                           const float* __restrict__ csum,
                           const float* __restrict__ cden,
                           float* __restrict__ X0) {
  const int c = blockIdx.x, h = blockIdx.y, v = threadIdx.x;
  float rn = csum[((size_t)c * kNH + h) * kDH + v];
  float rd = cden[(size_t)c * kNH + h];
  const int t0 = c * kChunk;
  for (int i = 0; i < kChunk; ++i) {
    int t = t0 + i;
    float a = Amat[(size_t)t * kNH + h];
    rd += a;
    rn += a * V[(size_t)t * kN + h * kDH + v];
    X0[(size_t)t * kN + h * kDH + v] = rn / rd;
  }
}

extern "C" void kernel_launch(void* const* d_in, const int* in_sizes, int n_in,
                              void* d_out, int out_size, void* d_ws, size_t ws_size,
                              hipStream_t stream) {
  const float* xs = (const float*)d_in[0];
  // d_in[1] (wq) is mathematically dead: the reference einsum contracts BOTH
  // d_head_k and d_head_q, so the Sum(act(q)) factor cancels in sq/zq exactly.
  const float* wk = (const float*)d_in[2];
  const float* wv = (const float*)d_in[3];
  const float* w1 = (const float*)d_in[4];
  const float* b1 = (const float*)d_in[5];
  const float* w2 = (const float*)d_in[6];
  const float* b2 = (const float*)d_in[7];
  float* out = (float*)d_out;

  float* ws = (float*)d_ws;
  size_t o = 0;
  float* V    = ws + o; o += (size_t)kSeq * kN;          // v projection
  float* Amat = ws + o; o += (size_t)kSeq * kNH;         // a_t[h] = sum_d act(k)
  float* csum = ws + o; o += (size_t)kNCh * kNH * kDH;   // chunk numerator sums
  float* cden = ws + o; o += (size_t)kNCh * kNH;         // chunk denominator sums
  float* X0   = ws + o; o += (size_t)kSeq * kN;          // attention output
  float* H    = ws + o; o += (size_t)kSeq * kN;          // mlp hidden

  // 2048 waves total: (8192/64) x (1024/64) tiles, 8 waves per 256-thread block
  dim3 gblk(256), gthr(256);

  gemm_f32_wmma<3><<<gblk, gthr, 0, stream>>>(xs, wk, nullptr, Amat); // k-proj + act + head reduce
  gemm_f32_wmma<0><<<gblk, gthr, 0, stream>>>(xs, wv, nullptr, V);    // v-proj
  scan_chunk_sums<<<dim3(kNCh, kNH), 64, 0, stream>>>(Amat, V, csum, cden);
  scan_prefix<<<1, 1024, 0, stream>>>(csum, cden);
  scan_apply<<<dim3(kNCh, kNH), 64, 0, stream>>>(Amat, V, csum, cden, X0);
  gemm_f32_wmma<1><<<gblk, gthr, 0, stream>>>(X0, w1, b1, H);         // mlp1 + bias + gelu
  gemm_f32_wmma<2><<<gblk, gthr, 0, stream>>>(H, w2, b2, out);        // mlp2 + bias
}